// SelfAttention_74397423501539
// MI455X (gfx1250) — compile-verified
//
#include <hip/hip_runtime.h>
#include <math.h>

// Problem constants (reference: B=32, DIM=4096, HQ=32, HK=8, HD=128, MAX_SEQ=2048)
#define Bsz     32
#define DIM     4096
#define HQ      32
#define HK      8
#define HD      128
#define MAX_SEQ 2048
#define NREP    (HQ / HK)   // 4

typedef __attribute__((ext_vector_type(2))) float v2f;
typedef __attribute__((ext_vector_type(8))) float v8f;

// D(16x16 f32) = A(16x4 f32) * B(4x16 f32) + C   -- full-precision matrix pipe
__device__ __forceinline__ v8f wmma4(v2f a, v2f b, v8f c) {
    return __builtin_amdgcn_wmma_f32_16x16x4_f32(
        /*neg_a=*/false, a, /*neg_b=*/false, b,
        /*c_mod=*/(short)0, c, /*reuse_a=*/false, /*reuse_b=*/false);
}

// ---------------------------------------------------------------------------
// Generic skinny GEMM: Y(32 x N) = A(32 x K) * W(K x N), row-major.
// One workgroup per 16-wide N tile; 8 waves split K; LDS tree reduction.
// A-frag layout (16x4 f32): lane L holds M=L%16, K = k0 + 2*(L/16) + {0,1}.
// B-frag layout (4x16 f32): lane L holds N=L%16, K = k0 + 2*(L/16) + {0,1}.
// D layout: lane<16 -> (M=v, N=lane); lane>=16 -> (M=8+v, N=lane-16).
// ---------------------------------------------------------------------------
__device__ void gemm32_tile(const float* __restrict__ A, int lda,
                            const float* __restrict__ W, int ldw,
                            float* __restrict__ Y, int ldy,
                            int K, int n0, float* red) {
    const int tid  = threadIdx.x;
    const int lane = tid & 31;
    const int wid  = tid >> 5;
    const int lrow = lane & 15;
    const int lhi  = lane >> 4;

    const int kper = K >> 3;        // K split across 8 waves
    const int kbeg = wid * kper;

    const float* a0p = A + (size_t)lrow * lda;          // M rows 0..15
    const float* a1p = A + (size_t)(16 + lrow) * lda;   // M rows 16..31
    const float* wp  = W + n0 + lrow;

    v8f acc0 = {}, acc1 = {};
#pragma unroll 4
    for (int k0 = kbeg; k0 < kbeg + kper; k0 += 4) {
        const int ka = k0 + 2 * lhi;
        v2f a0 = *(const v2f*)(a0p + ka);               // contiguous (k, k+1)
        v2f a1 = *(const v2f*)(a1p + ka);
        v2f b;
        b.x = wp[(size_t)ka * ldw];
        b.y = wp[(size_t)(ka + 1) * ldw];
        acc0 = wmma4(a0, b, acc0);
        acc1 = wmma4(a1, b, acc1);
    }

    // Per-wave partials -> LDS, then 256 threads reduce 8 partials each.
    float* my = red + wid * 512;
#pragma unroll
    for (int v = 0; v < 8; v++) {
        const int m = v + 8 * lhi;
        my[m * 16 + lrow]        = acc0[v];
        my[256 + m * 16 + lrow]  = acc1[v];
    }
    __syncthreads();
    for (int i = tid; i < 512; i += 256) {
        float s = 0.f;
#pragma unroll
        for (int w = 0; w < 8; w++) s += red[w * 512 + i];
        const int m = i >> 4, n = i & 15;
        Y[(size_t)m * ldy + n0 + n] = s;
    }
}

// 384 blocks: tiles [0,256) -> wq, [256,320) -> wk, [320,384) -> wv
__global__ __launch_bounds__(256) void qkv_gemm_kernel(
    const float* __restrict__ x,
    const float* __restrict__ wq, const float* __restrict__ wk,
    const float* __restrict__ wv,
    float* __restrict__ xq, float* __restrict__ xk, float* __restrict__ xv) {
    __shared__ float red[8 * 512];
    const int nt = blockIdx.x;
    if (nt < 256) {
        gemm32_tile(x, DIM, wq, HQ * HD, xq, HQ * HD, DIM, nt * 16, red);
    } else if (nt < 320) {
        gemm32_tile(x, DIM, wk, HK * HD, xk, HK * HD, DIM, (nt - 256) * 16, red);
    } else {
        gemm32_tile(x, DIM, wv, HK * HD, xv, HK * HD, DIM, (nt - 320) * 16, red);
    }
}

__global__ __launch_bounds__(256) void out_gemm_kernel(
    const float* __restrict__ attn, const float* __restrict__ wo,
    float* __restrict__ out) {
    __shared__ float red[8 * 512];
    gemm32_tile(attn, DIM, wo, DIM, out, DIM, DIM, blockIdx.x * 16, red);
}

// RoPE on xq (32x32x128) and xk (32x8x128); pair index i in [0,64)
__global__ __launch_bounds__(256) void rope_kernel(
    const float* __restrict__ cosv, const float* __restrict__ sinv,
    float* __restrict__ xq, float* __restrict__ xk) {
    const int t  = blockIdx.x * 256 + threadIdx.x;
    const int QP = Bsz * HQ * (HD / 2);   // 65536
    const int KP = Bsz * HK * (HD / 2);   // 16384
    if (t < QP) {
        const int i = t & 63;
        const int base = (t >> 6) * HD + 2 * i;
        const float c = cosv[i], s = sinv[i];
        const float tr = xq[base], ti = xq[base + 1];
        xq[base]     = tr * c - ti * s;
        xq[base + 1] = tr * s + ti * c;
    } else if (t < QP + KP) {
        const int u = t - QP;
        const int i = u & 63;
        const int base = (u >> 6) * HD + 2 * i;
        const float c = cosv[i], s = sinv[i];
        const float tr = xk[base], ti = xk[base + 1];
        xk[base]     = tr * c - ti * s;
        xk[base + 1] = tr * s + ti * c;
    }
}

// ---------------------------------------------------------------------------
// Attention: one workgroup per (b, g). 8 waves.
// Phase 1: scores(16seq x 16rows) = K_tile(16x4d) x q^T(4d x 16rows) via WMMA.
// Softmax over 2048 in LDS.  Phase 2: out(rows x d) = P(16x4s) x V(4s x 16d).
// s == start_pos uses freshly projected k/v (cache is an input; not mutated).
// ---------------------------------------------------------------------------
__global__ __launch_bounds__(256) void attn_kernel(
    const float* __restrict__ xq, const float* __restrict__ xk,
    const float* __restrict__ xv,
    const float* __restrict__ cache_k, const float* __restrict__ cache_v,
    const int* __restrict__ spp, float* __restrict__ attn) {
    __shared__ float sc[NREP][MAX_SEQ];     // 32 KB scores/probs
    __shared__ float qld[NREP][132];        // padded: avoids bank conflicts
    __shared__ float redm[NREP][64], reds[NREP][64];
    __shared__ float rowm[NREP], rowsum[NREP];

    const int tid  = threadIdx.x;
    const int lane = tid & 31;
    const int wid  = tid >> 5;
    const int lrow = lane & 15;
    const int lhi  = lane >> 4;
    const int b    = blockIdx.x >> 3;
    const int g    = blockIdx.x & 7;
    const int sp   = *spp;

    // Load the 4 q rows of this group into LDS (RoPE already applied).
    for (int i = tid; i < NREP * HD; i += 256) {
        const int r = i >> 7, d = i & (HD - 1);
        qld[r][d] = xq[(size_t)b * DIM + (g * NREP + r) * HD + d];
    }
    __syncthreads();

    // ---- Phase 1: scores ----
    const float scale = 0.08838834764831845f;   // 1/sqrt(128)
    const float* kbase = cache_k + ((size_t)b * MAX_SEQ * HK + g) * HD;
    const float* knew  = xk + ((size_t)b * HK + g) * HD;
    for (int t = wid; t < MAX_SEQ / 16; t += 8) {
        const int s = t * 16 + lrow;
        const float* arow = (s == sp) ? knew : (kbase + (size_t)s * (HK * HD));
        v8f acc = {};
#pragma unroll 8
        for (int d0 = 0; d0 < HD; d0 += 4) {
            v2f a = *(const v2f*)(arow + d0 + 2 * lhi);      // K[s][d], 8B load
            v2f bq;
            if (lrow < NREP) bq = *(const v2f*)&qld[lrow][d0 + 2 * lhi];
            else { bq.x = 0.f; bq.y = 0.f; }
            acc = wmma4(a, bq, acc);
        }
        if (lrow < NREP) {
#pragma unroll
            for (int v = 0; v < 8; v++) {
                const int sl = t * 16 + v + 8 * lhi;
                sc[lrow][sl] = (sl > sp) ? -INFINITY : acc[v] * scale;
            }
        }
    }
    __syncthreads();

    // ---- Softmax over seq (4 rows, 64 threads/row) ----
    {
        const int r = tid >> 6, c = tid & 63;
        float mx = -INFINITY;
        for (int s = c; s < MAX_SEQ; s += 64) mx = fmaxf(mx, sc[r][s]);
        redm[r][c] = mx;
        __syncthreads();
        if (c == 0) {
            float m = -INFINITY;
            for (int j = 0; j < 64; j++) m = fmaxf(m, redm[r][j]);
            rowm[r] = m;
        }
        __syncthreads();
        const float m = rowm[r];
        float sum = 0.f;
        for (int s = c; s < MAX_SEQ; s += 64) {
            const float p = __expf(sc[r][s] - m);   // exp(-inf)=0 handles mask
            sc[r][s] = p;
            sum += p;
        }
        reds[r][c] = sum;
        __syncthreads();
        if (c == 0) {
            float t2 = 0.f;
            for (int j = 0; j < 64; j++) t2 += reds[r][j];
            rowsum[r] = t2;
        }
        __syncthreads();
        const float inv = 1.f / rowsum[r];
        for (int s = c; s < MAX_SEQ; s += 64) sc[r][s] *= inv;
    }
    __syncthreads();

    // ---- Phase 2: out = P @ V ;  wave wid owns d-chunk [wid*16, wid*16+16) ----
    const float* vbase = cache_v + ((size_t)b * MAX_SEQ * HK + g) * HD;
    const float* vnew  = xv + ((size_t)b * HK + g) * HD;
    const int d = wid * 16 + lrow;
    v8f acc = {};
    for (int s0 = 0; s0 < MAX_SEQ; s0 += 4) {
        const int sa = s0 + 2 * lhi;
        v2f a;
        if (lrow < NREP) a = *(const v2f*)&sc[lrow][sa];    // contiguous probs
        else { a.x = 0.f; a.y = 0.f; }
        const float* v0 = (sa == sp)     ? vnew : (vbase + (size_t)sa * (HK * HD));
        const float* v1 = (sa + 1 == sp) ? vnew : (vbase + (size_t)(sa + 1) * (HK * HD));
        v2f bv;
        bv.x = v0[d];
        bv.y = v1[d];
        acc = wmma4(a, bv, acc);
    }
    // D rows 0..3 (q rows) live in lanes 0..15, VGPRs 0..3
    if (lhi == 0) {
#pragma unroll
        for (int v = 0; v < NREP; v++) {
            attn[(size_t)b * DIM + (g * NREP + v) * HD + d] = acc[v];
        }
    }
}

// ---------------------------------------------------------------------------
extern "C" void kernel_launch(void* const* d_in, const int* in_sizes, int n_in,
                              void* d_out, int out_size, void* d_ws, size_t ws_size,
                              hipStream_t stream) {
    (void)in_sizes; (void)n_in; (void)out_size; (void)ws_size;
    const float* x   = (const float*)d_in[0];
    const int*   spp = (const int*)d_in[1];
    const float* cs  = (const float*)d_in[2];
    const float* sn  = (const float*)d_in[3];
    const float* wq  = (const float*)d_in[4];
    const float* wk  = (const float*)d_in[5];
    const float* wv  = (const float*)d_in[6];
    const float* wo  = (const float*)d_in[7];
    const float* ck  = (const float*)d_in[8];
    const float* cv  = (const float*)d_in[9];
    float* out = (float*)d_out;

    // Workspace layout (f32): xq | xk | xv | attn  == 1.31 MB, fully rewritten
    float* xq = (float*)d_ws;
    float* xk = xq + (size_t)Bsz * HQ * HD;
    float* xv = xk + (size_t)Bsz * HK * HD;
    float* at = xv + (size_t)Bsz * HK * HD;

    qkv_gemm_kernel<<<dim3(384), dim3(256), 0, stream>>>(x, wq, wk, wv, xq, xk, xv);
    rope_kernel<<<dim3(320), dim3(256), 0, stream>>>(cs, sn, xq, xk);
    attn_kernel<<<dim3(Bsz * HK), dim3(256), 0, stream>>>(xq, xk, xv, ck, cv, spp, at);
    out_gemm_kernel<<<dim3(256), dim3(256), 0, stream>>>(at, wo, out);
}